// _RoIPooling_47983374630956
// MI455X (gfx1250) — compile-verified
//
#include <hip/hip_runtime.h>
#include <stdint.h>

// ---------------------------------------------------------------------------
// RoIPooling with context anchors (MI455X / gfx1250, wave32)
//
// Phase 1 (_ctx_anchor_kernel): one 256-thread block (8 wave32 waves) computes
//   the 8 context boxes per roi, the 32x256 IoU matrix, label gating and the
//   final cat_rois [N*9, 5], written straight into the tail of d_out.
// Phase 2 (_roipool_kernel): one thread per output element, coalesced stores.
//   Per-block box coords are broadcast via the CDNA5 async global->LDS path
//   (global_load_async_to_lds_b32 + s_wait_asynccnt), window rows prefetched
//   with global_prefetch_b8. Features (1.6 MB) live in L2 (192 MB).
// ---------------------------------------------------------------------------

#define PH 7
#define PW 7
#define FH 40
#define FW 40
#define SCALE 0.0625f

__global__ void _ctx_anchor_kernel(const float* __restrict__ rois,
                                   const int* __restrict__ hh_p,
                                   const int* __restrict__ hw_p,
                                   float* __restrict__ cat_out, int N) {
  __shared__ float sr[1024];              // rois staged in LDS, N*5 floats
  const int t = threadIdx.x;              // t in [0, N*8)
  if (t < N * 5) sr[t] = rois[t];
  __syncthreads();

  const float hh = (float)hh_p[0];
  const float hw = (float)hw_p[0];

  const int n = t >> 3;                   // roi index
  const int k = t & 7;                    // context cell 0..7
  const int m = (k < 4) ? k : (k + 1);    // skip center cell (index 4 of 9)

  const float x1 = sr[n * 5 + 1], y1 = sr[n * 5 + 2];
  const float x2 = sr[n * 5 + 3], y2 = sr[n * 5 + 4];
  const float w = x2 - x1, h = y2 - y1;

  // grid of shifted centers: cx from column (m%3), cy from row (m/3)
  const float cx = x1 + w * ((float)(m % 3) - 0.5f);
  const float cy = y1 + h * ((float)(m / 3) - 0.5f);
  float bx1 = cx - w * 0.25f, by1 = cy - h * 0.25f;
  float bx2 = cx + w * 0.25f, by2 = cy + h * 0.25f;
  const float bwv = bx2 - bx1 + 1.0f, bhv = by2 - by1 + 1.0f;
  const bool invalid = (bx1 < 0.0f) || (by1 < 0.0f) || (bx2 >= hw) ||
                       (by2 >= hh) || (bwv < 16.0f) || (bhv < 16.0f);
  if (invalid) { bx1 = x1; by1 = y1; bx2 = x2; by2 = y2; }

  // IoU of this context box against all N rois; first-max argmax.
  const float gw = bx2 - bx1 + 1.0f, gh = by2 - by1 + 1.0f;
  const float garea = gw * gh;
  float best_ov = -3.0e38f;
  int best = 0;
  for (int a = 0; a < N; ++a) {
    const float ax1 = sr[a * 5 + 1], ay1 = sr[a * 5 + 2];
    const float ax2 = sr[a * 5 + 3], ay2 = sr[a * 5 + 4];
    const float aw = ax2 - ax1 + 1.0f, ah = ay2 - ay1 + 1.0f;
    const float iw = fmaxf(fminf(ax2, bx2) - fmaxf(ax1, bx1) + 1.0f, 0.0f);
    const float ih = fmaxf(fminf(ay2, by2) - fmaxf(ay1, by1) + 1.0f, 0.0f);
    const float inter = iw * ih;
    float ov = inter / (aw * ah + garea - inter);
    if (gw == 1.0f && gh == 1.0f) ov = 0.0f;     // degenerate gt column
    if (aw == 1.0f && ah == 1.0f) ov = -1.0f;    // degenerate anchor row
    if (ov > best_ov) { best_ov = ov; best = a; }
  }

  bool label = (best_ov >= 0.3f);
  const float wc = bx2 - bx1, hc = by2 - by1;     // note: no +1 here (ref)
  const float rw = label ? (sr[best * 5 + 3] - sr[best * 5 + 1]) : 0.0f;
  const float rh = label ? (sr[best * 5 + 4] - sr[best * 5 + 2]) : 0.0f;
  label = label && !(fmaxf(rw, rh) >= fmaxf(wc, hc)) &&
          !(fminf(rw, rh) < fminf(wc, hc) / 3.0f);
  if (label) {
    bx1 = sr[best * 5 + 1]; by1 = sr[best * 5 + 2];
    bx2 = sr[best * 5 + 3]; by2 = sr[best * 5 + 4];
  }

  // cat_rois layout: row n*9 is the original roi, rows n*9+1..8 are contexts.
  float* o = cat_out + (size_t)(n * 9 + 1 + k) * 5;
  o[0] = 0.0f; o[1] = bx1; o[2] = by1; o[3] = bx2; o[4] = by2;
  if (k == 0) {
    float* r0 = cat_out + (size_t)(n * 9) * 5;
#pragma unroll
    for (int l = 0; l < 5; ++l) r0[l] = sr[n * 5 + l];
  }
}

__global__ void _roipool_kernel(const float* __restrict__ feat,
                                const float* __restrict__ cat,
                                float* __restrict__ out,
                                int C, int blocksPerBox) {
  // Single shared object -> guaranteed LDS offset 0 (target of async loads).
  __shared__ int smem32[8];

  const int b = blockIdx.x / blocksPerBox;           // box index
  const int sub = blockIdx.x - b * blocksPerBox;     // sub-block within box

  // CDNA5 async global->LDS broadcast of the 4 box coords (x1,y1,x2,y2):
  // lanes 0..3 of wave 0 each issue one B32 async copy into LDS offsets 0..12.
  if (threadIdx.x < 4) {
    unsigned int ldsoff = threadIdx.x * 4u;
    unsigned long long ga =
        (unsigned long long)(uintptr_t)(cat + (size_t)b * 5 + 1 + threadIdx.x);
    asm volatile("global_load_async_to_lds_b32 %0, %1, off"
                 :: "v"(ldsoff), "v"(ga)
                 : "memory");
  }
  asm volatile("s_wait_asynccnt 0" ::: "memory");   // ASYNCcnt split counter
  __syncthreads();

  const float x1 = __int_as_float(smem32[0]);
  const float y1 = __int_as_float(smem32[1]);
  const float x2 = __int_as_float(smem32[2]);
  const float y2 = __int_as_float(smem32[3]);

  // jnp.round == round-half-to-even == rintf (default RNE mode)
  const float sw = rintf(x1 * SCALE), sh = rintf(y1 * SCALE);
  const float ew = rintf(x2 * SCALE), eh = rintf(y2 * SCALE);
  const float bsh = fmaxf(eh - sh + 1.0f, 1.0f) / (float)PH;
  const float bsw = fmaxf(ew - sw + 1.0f, 1.0f) / (float)PW;

  const int lidx = sub * (int)blockDim.x + (int)threadIdx.x;  // c*49 + i*7 + j
  if (lidx >= C * PH * PW) return;
  const int c = lidx / (PH * PW);
  const int r = lidx - c * (PH * PW);
  const int i = r / PW;
  const int j = r - i * PW;

  const float hs = fminf(fmaxf(floorf((float)i * bsh) + sh, 0.0f), (float)FH);
  const float he = fminf(fmaxf(ceilf((float)(i + 1) * bsh) + sh, 0.0f), (float)FH);
  const float ws = fminf(fmaxf(floorf((float)j * bsw) + sw, 0.0f), (float)FW);
  const float we = fminf(fmaxf(ceilf((float)(j + 1) * bsw) + sw, 0.0f), (float)FW);
  const int h0 = (int)hs, h1 = (int)he, w0 = (int)ws, w1 = (int)we;

  const float* fc = feat + (size_t)c * (FH * FW);
  __builtin_prefetch(fc + h0 * FW + w0, 0, 1);      // global_prefetch_b8

  float mx = -1e30f;
  for (int y = h0; y < h1; ++y) {
    const float* rowp = fc + y * FW;
    for (int x = w0; x < w1; ++x) mx = fmaxf(mx, rowp[x]);
  }
  const float res = (h0 < h1 && w0 < w1) ? mx : 0.0f;

  out[(size_t)b * C * (PH * PW) + lidx] = res;      // fully coalesced store
}

extern "C" void kernel_launch(void* const* d_in, const int* in_sizes, int n_in,
                              void* d_out, int out_size, void* d_ws, size_t ws_size,
                              hipStream_t stream) {
  const float* features = (const float*)d_in[0];   // (1, C, 40, 40) f32
  const float* rois     = (const float*)d_in[1];   // (N, 5) f32
  const int*   hh       = (const int*)d_in[2];     // scalar
  const int*   hw       = (const int*)d_in[3];     // scalar

  const int N = in_sizes[1] / 5;                   // 32
  const int C = in_sizes[0] / (FH * FW);           // 256
  const int nBox = 9 * N;                          // 288

  float* out = (float*)d_out;
  // cat_rois lives at the tail of d_out (second tuple element), and is also
  // the input of the pooling kernel (same-stream ordering makes this safe).
  float* cat = out + (size_t)nBox * C * PH * PW;

  _ctx_anchor_kernel<<<1, N * 8, 0, stream>>>(rois, hh, hw, cat, N);

  const int bpb = (C * PH * PW + 255) / 256;       // 49 blocks per box (C=256)
  _roipool_kernel<<<nBox * bpb, 256, 0, stream>>>(features, cat, out, C, bpb);
}